// FullNet_29008209117719
// MI455X (gfx1250) — compile-verified
//
#include <hip/hip_runtime.h>
#include <math.h>

// tau[N,48,9] = 453 MB of f32 output -> store-bandwidth bound (~20us @ 23.3 TB/s).
// Phase 0: one tiny block precomputes Btab[432] = exp(w_lin)*filter into ws.
// Phase 1: WMMA-batched tiny MLPs (16 samples/wave, 1 gas/wave) -> ke[7,N] in ws.
// Phase 2: LDS-staged streaming writer, coalesced float4 (global_store_b128).

typedef float v2f __attribute__((ext_vector_type(2)));
typedef float v8f __attribute__((ext_vector_type(8)));

// 16-band filter masks (bit b = F16[g][b]); band b in [0,48) uses bit b/3.
__constant__ unsigned kMask[7] = {
    0xDFFFu, // h2o
    0x10D8u, // o3
    0x78FCu, // co2
    0x0401u, // o2
    0x5184u, // n2o
    0x8100u, // ch4
    0x1080u  // co
};

// ---------------- Phase 0: Btab[band*9+g] = exp(w_lin[g,band]) * filt ---------
__global__ __launch_bounds__(256) void btab_kernel(
    const float* __restrict__ w_lin, float* __restrict__ btG)
{
    for (int i = threadIdx.x; i < 432; i += 256) {   // i = band*9 + g
        int band = i / 9;
        int g    = i - band * 9;
        float coef = expf(w_lin[g * 48 + band]);
        float f = 1.0f;
        if (g >= 2) f = (float)((kMask[g - 2] >> (band / 3)) & 1u);
        btG[i] = coef * f;
    }
}

// ---------------- Phase 1: ke = sigmoid(MLP(t_p)) per gas, 16 samples/wave ----
__global__ __launch_bounds__(256) void ke_kernel(
    const float* __restrict__ t_p,
    const float* __restrict__ W1, const float* __restrict__ b1,
    const float* __restrict__ W2, const float* __restrict__ b2,
    const float* __restrict__ W3, const float* __restrict__ b3,
    const float* __restrict__ Wo, const float* __restrict__ bo,
    float* __restrict__ ke_out, int nTiles, int N)
{
    const int wid  = blockIdx.x * 8 + (threadIdx.x >> 5);
    const int lane = threadIdx.x & 31;
    // gas/tile are wave-uniform; force SGPRs so weight fetches become s_loads.
    const int gas  = __builtin_amdgcn_readfirstlane(wid / nTiles);
    const int tile = __builtin_amdgcn_readfirstlane(wid - gas * nTiles);
    if (gas >= 7) return; // whole wave exits together

    const int  m  = lane & 15;
    const bool lo = lane < 16;
    const int  n  = tile * 16 + m;

    // ---- Layer 1: h1 = relu(W1[6x2]*[t;p] + b1) via one 16x16x4 f32 WMMA.
    // A (16x4): col0=W1[:,0] col1=W1[:,1] col2=b1 col3=0 ; B rows: [t; p; 1; 0]
    const int  mc6 = m < 6 ? m : 5;
    const bool mv6 = m < 6;
    float w10 = W1[gas * 12 + mc6 * 2 + 0];
    float w11 = W1[gas * 12 + mc6 * 2 + 1];
    float b1v = b1[gas * 6 + mc6];
    v2f a1v;
    a1v[0] = mv6 ? (lo ? w10 : b1v) : 0.0f;   // lanes0-15: K0 ; lanes16-31: K2
    a1v[1] = (lo && mv6) ? w11 : 0.0f;        // lanes0-15: K1 ; lanes16-31: K3=0
    float tt = t_p[2 * n + 0];
    float pp = t_p[2 * n + 1];
    v2f bb;
    bb[0] = lo ? tt : 1.0f;
    bb[1] = lo ? pp : 0.0f;
    v8f acc1 = {0.f,0.f,0.f,0.f,0.f,0.f,0.f,0.f};
    acc1 = __builtin_amdgcn_wmma_f32_16x16x4_f32(false, a1v, false, bb,
                                                 (short)0, acc1, false, false);
    #pragma unroll
    for (int i = 0; i < 8; ++i) acc1[i] = fmaxf(acc1[i], 0.0f);

    // ---- Layer 2: h2 = relu(W2[4x6]*h1 + b2) -> two accumulating WMMAs (K 0..3, 4..5)
    const int  mc4 = m < 4 ? m : 3;
    const bool mv4 = m < 4;
    const float* W2g = W2 + gas * 24 + mc4 * 6;
    float w20 = W2g[0], w21 = W2g[1], w22 = W2g[2];
    float w23 = W2g[3], w24 = W2g[4], w25 = W2g[5];
    // Hoist bias loads unconditionally (scalar addresses) -> select via cndmask.
    float b20 = b2[gas * 4 + 0], b21 = b2[gas * 4 + 1];
    float b22 = b2[gas * 4 + 2], b23 = b2[gas * 4 + 3];
    v2f a2a;
    a2a[0] = mv4 ? (lo ? w20 : w22) : 0.0f;
    a2a[1] = mv4 ? (lo ? w21 : w23) : 0.0f;
    float r2 = __shfl(acc1[2], lane & 15, 32);  // h1 row2 -> lanes 16-31 (K=2)
    float r3 = __shfl(acc1[3], lane & 15, 32);  // h1 row3 -> lanes 16-31 (K=3)
    v2f b2a;
    b2a[0] = lo ? acc1[0] : r2;
    b2a[1] = lo ? acc1[1] : r3;
    v8f acc2 = {0.f,0.f,0.f,0.f,0.f,0.f,0.f,0.f};
    acc2[0] = lo ? b20 : 0.0f;                 // bias in C operand, rows 0..3
    acc2[1] = lo ? b21 : 0.0f;
    acc2[2] = lo ? b22 : 0.0f;
    acc2[3] = lo ? b23 : 0.0f;
    acc2 = __builtin_amdgcn_wmma_f32_16x16x4_f32(false, a2a, false, b2a,
                                                 (short)0, acc2, false, false);
    v2f a2b;
    a2b[0] = (lo && mv4) ? w24 : 0.0f;         // K0 = h-index 4; K2/K3 = 0
    a2b[1] = (lo && mv4) ? w25 : 0.0f;         // K1 = h-index 5
    v2f b2b;
    b2b[0] = lo ? acc1[4] : 0.0f;              // h1 row4 ; row6 = 0
    b2b[1] = lo ? acc1[5] : 0.0f;              // h1 row5 ; row7 = 0
    acc2 = __builtin_amdgcn_wmma_f32_16x16x4_f32(false, a2b, false, b2b,
                                                 (short)0, acc2, false, false);
    #pragma unroll
    for (int i = 0; i < 8; ++i) acc2[i] = fmaxf(acc2[i], 0.0f);

    // ---- Layer 3: h3 = relu(W3[4x4]*h2 + b3) -> one WMMA
    const float* W3g = W3 + gas * 16 + mc4 * 4;
    float w30 = W3g[0], w31 = W3g[1], w32 = W3g[2], w33 = W3g[3];
    float b30 = b3[gas * 4 + 0], b31 = b3[gas * 4 + 1];
    float b32v = b3[gas * 4 + 2], b33 = b3[gas * 4 + 3];
    v2f a3;
    a3[0] = mv4 ? (lo ? w30 : w32) : 0.0f;
    a3[1] = mv4 ? (lo ? w31 : w33) : 0.0f;
    float s2 = __shfl(acc2[2], lane & 15, 32);
    float s3 = __shfl(acc2[3], lane & 15, 32);
    v2f b3v;
    b3v[0] = lo ? acc2[0] : s2;
    b3v[1] = lo ? acc2[1] : s3;
    v8f acc3 = {0.f,0.f,0.f,0.f,0.f,0.f,0.f,0.f};
    acc3[0] = lo ? b30 : 0.0f;
    acc3[1] = lo ? b31 : 0.0f;
    acc3[2] = lo ? b32v : 0.0f;
    acc3[3] = lo ? b33 : 0.0f;
    acc3 = __builtin_amdgcn_wmma_f32_16x16x4_f32(false, a3, false, b3v,
                                                 (short)0, acc3, false, false);
    #pragma unroll
    for (int i = 0; i < 4; ++i) acc3[i] = fmaxf(acc3[i], 0.0f);

    // ---- Output: ke = sigmoid(Wo . h3 + bo); lane k holds sample column k.
    float wo0 = Wo[gas * 4 + 0], wo1 = Wo[gas * 4 + 1];
    float wo2 = Wo[gas * 4 + 2], wo3 = Wo[gas * 4 + 3];
    float bov = bo[gas];
    float logit = wo0 * acc3[0] + wo1 * acc3[1] + wo2 * acc3[2] + wo3 * acc3[3] + bov;
    float kev = 1.0f / (1.0f + expf(-logit));
    if (lo) ke_out[gas * N + n] = kev;
}

// ---------------- Phase 2: tau[n,band,g] = Btab[band*9+g] * m[n][g] -----------
// 64 samples/block; Btab (432) + m (64*9) staged in LDS; coalesced float4 stores.
__global__ __launch_bounds__(256) void tau_kernel(
    const float* __restrict__ c, const float* __restrict__ btG,
    const float* __restrict__ ke, float* __restrict__ out, int N)
{
    __shared__ float Bt[432];
    __shared__ float Ms[576];
    const int tid = threadIdx.x;
    const int n0  = blockIdx.x * 64;

    for (int i = tid; i < 432; i += 256)            // 1.7 KB, L2-resident
        Bt[i] = btG[i];
    for (int p = tid; p < 576; p += 256) {          // p = s*9 + g ; contiguous c read
        int s = p / 9;
        int g = p - s * 9;
        float val = c[n0 * 9 + p];
        int gg = g >= 2 ? g - 2 : 0;
        float kv = ke[gg * N + n0 + s];
        Ms[p] = val * (g >= 2 ? kv : 1.0f);
    }
    __syncthreads();

    float4* outv = (float4*)out + (size_t)blockIdx.x * 6912; // 64*432/4
    #pragma unroll 1
    for (int it = 0; it < 27; ++it) {
        int q  = it * 256 + tid;                    // float4 index in tile
        int s  = q / 108;                           // sample within tile (432/4=108)
        int j4 = (q - s * 108) * 4;                 // float index within sample row
        int g0 = j4 % 9;
        int g1 = g0 + 1; g1 = (g1 == 9) ? 0 : g1;
        int g2 = g1 + 1; g2 = (g2 == 9) ? 0 : g2;
        int g3 = g2 + 1; g3 = (g3 == 9) ? 0 : g3;
        int base = s * 9;
        float4 v;
        v.x = Bt[j4 + 0] * Ms[base + g0];
        v.y = Bt[j4 + 1] * Ms[base + g1];
        v.z = Bt[j4 + 2] * Ms[base + g2];
        v.w = Bt[j4 + 3] * Ms[base + g3];
        outv[q] = v;
    }
}

extern "C" void kernel_launch(void* const* d_in, const int* in_sizes, int n_in,
                              void* d_out, int out_size, void* d_ws, size_t ws_size,
                              hipStream_t stream) {
    const float* t_p   = (const float*)d_in[0];
    const float* c     = (const float*)d_in[1];
    const float* w_lin = (const float*)d_in[2];
    const float* W1    = (const float*)d_in[3];
    const float* b1    = (const float*)d_in[4];
    const float* W2    = (const float*)d_in[5];
    const float* b2    = (const float*)d_in[6];
    const float* W3    = (const float*)d_in[7];
    const float* b3    = (const float*)d_in[8];
    const float* Wo    = (const float*)d_in[9];
    const float* bo    = (const float*)d_in[10];

    const int N = in_sizes[0] / 2;          // t_p is [N,2]
    float* ke_ws = (float*)d_ws;            // 7*N floats
    float* btG   = ke_ws + (size_t)7 * N;   // + 432 floats

    btab_kernel<<<1, 256, 0, stream>>>(w_lin, btG);

    const int nTiles  = N / 16;             // 16 samples per wave
    const int blocksA = (7 * nTiles + 7) / 8;
    ke_kernel<<<blocksA, 256, 0, stream>>>(t_p, W1, b1, W2, b2, W3, b3, Wo, bo,
                                           ke_ws, nTiles, N);

    const int blocksB = N / 64;             // 64 samples per block
    tau_kernel<<<blocksB, 256, 0, stream>>>(c, btG, ke_ws, (float*)d_out, N);
}